// SinkhornProjection_85392539779779
// MI455X (gfx1250) — compile-verified
//
#include <hip/hip_runtime.h>
#include <math.h>

// Sinkhorn projection, separable form: P = diag(u) * exp(M) * diag(v).
// 10 iterations reduce to 20 matvec passes over M + one final elementwise pass.
// HBM traffic ~ 21 reads of M (512MB) + 1 write of P (512MB) ~ 11 GB -> ~0.47ms @ 23.3 TB/s.
// Row-sum reduction uses V_WMMA_F32_16X16X4_F32 with a ones B-matrix (exact f32 FMA,
// layout-invariant B); A and C/D VGPR layouts per CDNA5 ISA.
//
// d_ws layout (floats): u[32*2048], v[32*2048], tpart[32*8*2048]  => 2,621,440 bytes.

#define NB 32
#define NN 2048
#define EPSV 1e-8f
#define TILE_ROWS 16
#define CHUNK 512
#define LDS_STRIDE 516   // 512 + 4 pad: float4-aligned stores, conflict-free WMMA-A reads
#define COL_GROUPS 8
#define COL_ROWS 256     // rows per column-pass partial group

typedef __attribute__((ext_vector_type(2))) float v2f;
typedef __attribute__((ext_vector_type(8))) float v8f;

// ---------------------------------------------------------------------------
// init: u = v = 1
__global__ __launch_bounds__(256) void sink_init(float* __restrict__ u,
                                                 float* __restrict__ v) {
    int idx = blockIdx.x * 256 + threadIdx.x;
    u[idx] = 1.0f;
    v[idx] = 1.0f;
}

// ---------------------------------------------------------------------------
// Row pass: s_i = sum_j exp(M[b][i][j]) * v[b][j];  u_i <- u_i / (u_i*s_i + eps)
// One block = 16 rows of one batch slice. 8 waves; wave w owns 64 columns of
// each 512-wide chunk. WMMA (A=data tile 16x4, B=ones) accumulates row sums.
__global__ __launch_bounds__(256) void sink_row(const float* __restrict__ M,
                                                float* __restrict__ u,
                                                const float* __restrict__ v) {
    __shared__ float tile[TILE_ROWS * LDS_STRIDE];
    __shared__ float part[TILE_ROWS][8];

    const int b    = blockIdx.y;
    const int i0   = blockIdx.x * TILE_ROWS;
    const int tid  = threadIdx.x;
    const int lane = tid & 31;
    const int wave = tid >> 5;

    const float* Mrow = M + ((size_t)b * NN + i0) * NN;
    const float* vb   = v + b * NN;

    v8f acc = {};                       // per-wave row-sum partials (all D cols equal)
    const v2f ones = {1.0f, 1.0f};

    for (int j0 = 0; j0 < NN; j0 += CHUNK) {
        // ---- coalesced load: 16 rows x 512 cols = 2048 float4, 8 per thread ----
#pragma unroll
        for (int k = 0; k < 8; ++k) {
            int e4  = tid + k * 256;          // float4 index inside the tile
            int row = e4 >> 7;                // / (512/4)
            int col = (e4 & 127) << 2;
            const float4 m  = *(const float4*)(Mrow + (size_t)row * NN + j0 + col);
            const float4 vv = *(const float4*)(vb + j0 + col);
            float4 r;
            r.x = __expf(m.x) * vv.x;
            r.y = __expf(m.y) * vv.y;
            r.z = __expf(m.z) * vv.z;
            r.w = __expf(m.w) * vv.w;
            *(float4*)(&tile[row * LDS_STRIDE + col]) = r;
        }
        __syncthreads();

        // ---- WMMA reduction: wave w covers cols [64w, 64w+64) in steps of K=4 ----
        const int rrow = lane & 15;           // A-matrix M index = lane % 16
        const int koff = (lane >> 4) << 1;    // lanes 16-31 hold K=2,3
        const int base = wave * 64;
#pragma unroll
        for (int kk = 0; kk < 16; ++kk) {
            const int c = base + kk * 4;
            const float* p = &tile[rrow * LDS_STRIDE + c + koff];
            v2f a;
            a.x = p[0];
            a.y = p[1];
            acc = __builtin_amdgcn_wmma_f32_16x16x4_f32(
                false, a, false, ones, (short)0, acc, false, false);
        }
        __syncthreads();
    }

    // D layout: VGPR p -> M=p (lanes 0-15), M=p+8 (lanes 16-31); cols identical.
    if (lane == 0) {
#pragma unroll
        for (int p = 0; p < 8; ++p) part[p][wave] = acc[p];
    } else if (lane == 16) {
#pragma unroll
        for (int p = 0; p < 8; ++p) part[p + 8][wave] = acc[p];
    }
    __syncthreads();

    if (tid < TILE_ROWS) {
        float s = 0.0f;
#pragma unroll
        for (int w = 0; w < 8; ++w) s += part[tid][w];
        const int gi = b * NN + i0 + tid;
        const float uo = u[gi];
        u[gi] = uo / (uo * s + EPSV);
    }
}

// ---------------------------------------------------------------------------
// Column pass partials: tpart[b][g][j] = sum_{i in group g} exp(M[b][i][j]) * u[b][i]
// Block = 256 consecutive columns x 256 rows. Fully coalesced; no atomics.
__global__ __launch_bounds__(256) void sink_col(const float* __restrict__ M,
                                                const float* __restrict__ u,
                                                float* __restrict__ tpart) {
    __shared__ float su[COL_ROWS];
    const int b  = blockIdx.z;
    const int g  = blockIdx.y;
    const int j  = blockIdx.x * 256 + threadIdx.x;
    const int i0 = g * COL_ROWS;

    su[threadIdx.x] = u[b * NN + i0 + threadIdx.x];
    __syncthreads();

    const float* Mb = M + (size_t)b * NN * NN;
    float acc = 0.0f;
#pragma unroll 4
    for (int ii = 0; ii < COL_ROWS; ++ii) {
        acc += __expf(Mb[(size_t)(i0 + ii) * NN + j]) * su[ii];
    }
    tpart[((size_t)(b * COL_GROUPS + g)) * NN + j] = acc;
}

// ---------------------------------------------------------------------------
// Reduce column partials (fixed order -> deterministic), update v.
__global__ __launch_bounds__(256) void sink_vupd(const float* __restrict__ tpart,
                                                 float* __restrict__ v) {
    const int b = blockIdx.y;
    const int j = blockIdx.x * 256 + threadIdx.x;
    float t = 0.0f;
#pragma unroll
    for (int g = 0; g < COL_GROUPS; ++g)
        t += tpart[((size_t)(b * COL_GROUPS + g)) * NN + j];
    const int gi = b * NN + j;
    const float vo = v[gi];
    v[gi] = vo / (vo * t + EPSV);
}

// ---------------------------------------------------------------------------
// Final: P[b][i][j] = u[b][i] * exp(M[b][i][j]) * v[b][j]   (float4 vectorized)
__global__ __launch_bounds__(256) void sink_final(const float* __restrict__ M,
                                                  const float* __restrict__ u,
                                                  const float* __restrict__ v,
                                                  float* __restrict__ out) {
    const size_t idx4 = (size_t)blockIdx.x * 256 + threadIdx.x;
    const size_t e    = idx4 * 4;
    const int b   = (int)(e >> 22);            // / (2048*2048)
    const int rem = (int)(e & (size_t)(NN * NN - 1));
    const int i   = rem >> 11;                 // / 2048
    const int j   = rem & (NN - 1);

    const float4 m  = ((const float4*)M)[idx4];
    const float4 vv = *(const float4*)(v + b * NN + j);
    const float  ui = u[b * NN + i];

    float4 o;
    o.x = ui * __expf(m.x) * vv.x;
    o.y = ui * __expf(m.y) * vv.y;
    o.z = ui * __expf(m.z) * vv.z;
    o.w = ui * __expf(m.w) * vv.w;
    ((float4*)out)[idx4] = o;
}

// ---------------------------------------------------------------------------
extern "C" void kernel_launch(void* const* d_in, const int* in_sizes, int n_in,
                              void* d_out, int out_size, void* d_ws, size_t ws_size,
                              hipStream_t stream) {
    const float* M  = (const float*)d_in[0];
    float* out      = (float*)d_out;
    float* u        = (float*)d_ws;                 // 32*2048 floats
    float* v        = u + NB * NN;                  // 32*2048 floats
    float* tpart    = v + NB * NN;                  // 32*8*2048 floats
    (void)in_sizes; (void)n_in; (void)out_size; (void)ws_size;

    sink_init<<<(NB * NN) / 256, 256, 0, stream>>>(u, v);

    for (int it = 0; it < 10; ++it) {
        sink_row<<<dim3(NN / TILE_ROWS, NB), 256, 0, stream>>>(M, u, v);
        sink_col<<<dim3(NN / 256, COL_GROUPS, NB), 256, 0, stream>>>(M, u, tpart);
        sink_vupd<<<dim3(NN / 256, NB), 256, 0, stream>>>(tpart, v);
    }

    const size_t total4 = (size_t)NB * NN * NN / 4;
    sink_final<<<(unsigned)(total4 / 256), 256, 0, stream>>>(M, u, v, out);
}